// SoftRankLoss_28132035789055
// MI455X (gfx1250) — compile-verified
//
#include <hip/hip_runtime.h>

// ---------------------------------------------------------------------------
// log|det(F^T F + delta*I)| summed over 4M 3x3 batches.
// Bandwidth-bound (144 MB in / 4 B out, ~1.5 flop/byte): stream via CDNA5
// async global->LDS copies (b128: 512 B dense per wave-instruction), VALU
// compute, deterministic tree reduction. No WMMA on purpose: 3x3 tiles waste
// >80% of a 16x16 WMMA and the kernel is HBM-bound at 23.3 TB/s anyway.
// ---------------------------------------------------------------------------

#define TPB 256
#define DELTA_REG 1e-6f

#if defined(__HIP_DEVICE_COMPILE__) && defined(__gfx1250__)
#define ON_GFX1250 1
#else
#define ON_GFX1250 0
#endif

#if ON_GFX1250 && __has_builtin(__builtin_amdgcn_global_load_async_to_lds_b32)
#define HAVE_ASYNC_LDS 1
#else
#define HAVE_ASYNC_LDS 0
#endif

#if HAVE_ASYNC_LDS && __has_builtin(__builtin_amdgcn_global_load_async_to_lds_b128)
#define HAVE_ASYNC_LDS_B128 1
#else
#define HAVE_ASYNC_LDS_B128 0
#endif

typedef int v4i __attribute__((ext_vector_type(4)));
typedef __attribute__((address_space(1))) int gas_int;   // global (AS1)
typedef __attribute__((address_space(3))) int las_int;   // LDS (AS3)
typedef __attribute__((address_space(1))) v4i gas_v4i;
typedef __attribute__((address_space(3))) v4i las_v4i;

// f[] is one 3x3 matrix, row-major: f[i*3+j] = F[i][j].
// FtF[j][k] = sum_i F[i][j]*F[i][k]  (symmetric; 6 unique entries)
__device__ __forceinline__ float logabsdet_ftf(const float f[9]) {
  const float c00 = f[0], c01 = f[1], c02 = f[2];
  const float c10 = f[3], c11 = f[4], c12 = f[5];
  const float c20 = f[6], c21 = f[7], c22 = f[8];

  const float a = fmaf(c00, c00, fmaf(c10, c10, c20 * c20)) + DELTA_REG; // (0,0)
  const float b = fmaf(c00, c01, fmaf(c10, c11, c20 * c21));             // (0,1)
  const float c = fmaf(c00, c02, fmaf(c10, c12, c20 * c22));             // (0,2)
  const float d = fmaf(c01, c01, fmaf(c11, c11, c21 * c21)) + DELTA_REG; // (1,1)
  const float e = fmaf(c01, c02, fmaf(c11, c12, c21 * c22));             // (1,2)
  const float g = fmaf(c02, c02, fmaf(c12, c12, c22 * c22)) + DELTA_REG; // (2,2)

  // det of symmetric [a b c; b d e; c e g]
  const float m0 = fmaf(d, g, -(e * e));
  const float m1 = fmaf(b, g, -(c * e));
  const float m2 = fmaf(b, e, -(c * d));
  const float det = fmaf(a, m0, fmaf(-b, m1, c * m2));
  return logf(fabsf(det));
}

__global__ __launch_bounds__(TPB) void logdet_partial_kernel(
    const float* __restrict__ Fm, float* __restrict__ partial, int nmat) {
  __shared__ float tile[TPB * 9];        // 9216 B staging tile (576 float4)
  __shared__ float wavesum[TPB / 32];    // 8 waves per block

  const int tid    = threadIdx.x;
  const int mbase  = blockIdx.x * TPB;   // first matrix of tile  (< 4M)
  const int ebase  = mbase * 9;          // first float of tile   (< 36M)
  const int etotal = nmat * 9;
  const bool full_tile = (mbase + TPB) <= nmat;  // uniform branch

#if HAVE_ASYNC_LDS
  if (full_tile) {
#if HAVE_ASYNC_LDS_B128
    // 576 dense float4 copies: 2 rounds of 256 lanes + 1 round of 64 lanes.
    // Each wave-instruction streams 512 B global -> LDS, no VGPR round trip.
    gas_v4i* gsrc = (gas_v4i*)(Fm + ebase);
    las_v4i* ldst = (las_v4i*)tile;
#pragma unroll
    for (int k = 0; k < 2; ++k)
      __builtin_amdgcn_global_load_async_to_lds_b128(
          gsrc + k * TPB + tid, ldst + k * TPB + tid, 0, 0);
    if (tid < TPB / 4)
      __builtin_amdgcn_global_load_async_to_lds_b128(
          gsrc + 2 * TPB + tid, ldst + 2 * TPB + tid, 0, 0);
#else
#pragma unroll
    for (int k = 0; k < 9; ++k)
      __builtin_amdgcn_global_load_async_to_lds_b32(
          (gas_int*)(Fm + ebase + k * TPB + tid),
          (las_int*)&tile[k * TPB + tid], 0, 0);
#endif
  } else {
    // Generic tail tile: per-element guard, 32-bit indices.
#pragma unroll
    for (int k = 0; k < 9; ++k) {
      const int idx = ebase + k * TPB + tid;
      if (idx < etotal)
        __builtin_amdgcn_global_load_async_to_lds_b32(
            (gas_int*)(Fm + idx), (las_int*)&tile[k * TPB + tid], 0, 0);
    }
  }
#if __has_builtin(__builtin_amdgcn_s_wait_asynccnt)
  __builtin_amdgcn_s_wait_asynccnt(0);
#else
  asm volatile("s_wait_asynccnt 0" ::: "memory");
#endif
  __syncthreads();  // each wave waited on its own copies; barrier publishes
                    // the tile to all 8 waves
#else
  // Fallback: plain coalesced loads staged through VGPRs into LDS.
#pragma unroll
  for (int k = 0; k < 9; ++k) {
    const int idx = ebase + k * TPB + tid;
    tile[k * TPB + tid] = (idx < etotal) ? Fm[idx] : 0.0f;
  }
  __syncthreads();
#endif

  // Strided LDS read, stride 9 floats: gcd(9,64)=1 -> bank-conflict-free.
  float v = 0.0f;
  if (mbase + tid < nmat) {
    float f[9];
#pragma unroll
    for (int i = 0; i < 9; ++i) f[i] = tile[tid * 9 + i];
    v = logabsdet_ftf(f);
  }

  // Deterministic reduction: intra-wave xor-shuffle tree (wave32), then
  // cross-wave tree via LDS. Fixed pairing order -> bit-stable across replays.
#pragma unroll
  for (int off = 16; off > 0; off >>= 1) v += __shfl_xor(v, off, 32);
  const int wave = tid >> 5;
  const int lane = tid & 31;
  if (lane == 0) wavesum[wave] = v;
  __syncthreads();
  if (wave == 0) {
    float s = (lane < (TPB / 32)) ? wavesum[lane] : 0.0f;
#pragma unroll
    for (int off = 16; off > 0; off >>= 1) s += __shfl_xor(s, off, 32);
    if (lane == 0) partial[blockIdx.x] = s;
  }
}

__global__ __launch_bounds__(TPB) void final_reduce_kernel(
    const float* __restrict__ partial, int n, float* __restrict__ out) {
  __shared__ float wavesum[TPB / 32];
  const int tid = threadIdx.x;
  float s = 0.0f;
  for (int i = tid; i < n; i += TPB) s += partial[i];  // fixed order
#pragma unroll
  for (int off = 16; off > 0; off >>= 1) s += __shfl_xor(s, off, 32);
  const int wave = tid >> 5;
  const int lane = tid & 31;
  if (lane == 0) wavesum[wave] = s;
  __syncthreads();
  if (wave == 0) {
    float t = (lane < (TPB / 32)) ? wavesum[lane] : 0.0f;
#pragma unroll
    for (int off = 16; off > 0; off >>= 1) t += __shfl_xor(t, off, 32);
    if (lane == 0) out[0] = t;
  }
}

extern "C" void kernel_launch(void* const* d_in, const int* in_sizes, int n_in,
                              void* d_out, int out_size, void* d_ws, size_t ws_size,
                              hipStream_t stream) {
  (void)n_in; (void)out_size; (void)ws_size;
  const float* Fm  = (const float*)d_in[0];
  float* out       = (float*)d_out;
  float* partial   = (float*)d_ws;

  const int nmat    = in_sizes[0] / 9;              // 4,000,000
  const int nblocks = (nmat + TPB - 1) / TPB;       // 15,625 tiles

  logdet_partial_kernel<<<nblocks, TPB, 0, stream>>>(Fm, partial, nmat);
  final_reduce_kernel<<<1, TPB, 0, stream>>>(partial, nblocks, out);
}